// WDMDemux_4200478015872
// MI455X (gfx1250) — compile-verified
//
#include <hip/hip_runtime.h>
#include <stdint.h>

typedef __attribute__((ext_vector_type(2))) float v2f;
typedef __attribute__((ext_vector_type(8))) float v8f;
typedef __attribute__((ext_vector_type(4))) unsigned int u32x4;
typedef __attribute__((ext_vector_type(8))) int i32x8;
typedef __attribute__((ext_vector_type(4))) int i32x4;

#define TILE_T    256                 // output times per workgroup
#define HALO_M    1024                // FIR taps per phase
#define M_SPAN    (HALO_M + TILE_T)   // 1280 polyphase columns staged
#define SIG_STAGE (M_SPAN * 16)       // 20480 floats per (re|im)

// ---- Tensor Data Mover: 1-D contiguous load of nelem f32 into LDS ----
__device__ __forceinline__ void tdm_load_1d(unsigned lds_byte,
                                            unsigned long long gaddr,
                                            unsigned nelem) {
  u32x4 g0;
  g0.x = 1u;                                                      // count=1 valid D#
  g0.y = lds_byte;                                                // lds_addr (bytes)
  g0.z = (unsigned)(gaddr & 0xFFFFFFFFull);                       // global_addr lo
  g0.w = (unsigned)((gaddr >> 32) & 0x01FFFFFFull) | (2u << 30);  // addr hi | type=2
  i32x8 g1;
  g1[0] = 0x00020000;                            // data_size=2 (4B), wg_mask=0
  g1[1] = (int)((nelem & 0xFFFFu) << 16);        // tensor_dim0[15:0]
  g1[2] = (int)(((nelem >> 16) & 0xFFFFu) | (1u << 16)); // dim0[31:16] | tensor_dim1=1
  g1[3] = (int)((nelem & 0xFFFFu) << 16);        // tile_dim0 = nelem
  g1[4] = 0;                                     // tile_dim1/2 unused
  g1[5] = (int)nelem;                            // tensor_dim0_stride lo32
  g1[6] = 0;
  g1[7] = 0;
  i32x4 z4 = (i32x4)0;
#if defined(__clang_major__) && (__clang_major__ >= 23)
  i32x8 z8 = (i32x8)0;
  __builtin_amdgcn_tensor_load_to_lds(g0, g1, z4, z4, z8, 0);
#else
  __builtin_amdgcn_tensor_load_to_lds(g0, g1, z4, z4, 0);
#endif
}

// One 16-tap shift block: 4 K-chunks x (re,im) x 2 rows = 16 WMMAs.
// MODE 0: clamp taps >1023 (eidx==0); MODE 1: interior (no mask);
// MODE 2: clamp taps <0 (eidx==64).
template<int MODE>
__device__ __forceinline__ void fir_step(const float* __restrict__ sc,
                                         const float* __restrict__ ss,
                                         int kbase, int baseB,
                                         int b, int hl, int r0,
                                         v8f& re0, v8f& im0, v8f& re1, v8f& im1) {
  #pragma unroll
  for (int chunk = 0; chunk < 4; ++chunk) {
    v2f a0, a1, br0, bi0, br1, bi1;
    #pragma unroll
    for (int i = 0; i < 2; ++i) {
      const int d = (chunk << 2) + (hl << 1) + i;   // K index 0..15
      int k = b - d + kbase;                        // filter tap
      float m = 1.f;
      if (MODE == 0) { if (k > 1023) { m = 0.f; k = 1023; } }
      if (MODE == 2) { if (k < 0)    { m = 0.f; k = 0; } }
      // rows r0 (even) and r0+1 are adjacent -> one b64 per pair
      const float2 av = *(const float2*)&sc[(k << 4) + r0];
      a0[i] = (MODE == 1) ? av.x : av.x * m;
      a1[i] = (MODE == 1) ? av.y : av.y * m;
      const int offB = baseB + (d << 4);            // even float index
      const float2 vr = *(const float2*)&ss[offB];            // {15-r1, 15-r0}
      br1[i] = vr.x; br0[i] = vr.y;
      const float2 vi = *(const float2*)&ss[SIG_STAGE + offB];
      bi1[i] = vi.x; bi0[i] = vi.y;
    }
    re0 = __builtin_amdgcn_wmma_f32_16x16x4_f32(false, a0, false, br0, (short)0, re0, false, false);
    im0 = __builtin_amdgcn_wmma_f32_16x16x4_f32(false, a0, false, bi0, (short)0, im0, false, false);
    re1 = __builtin_amdgcn_wmma_f32_16x16x4_f32(false, a1, false, br1, (short)0, re1, false, false);
    im1 = __builtin_amdgcn_wmma_f32_16x16x4_f32(false, a1, false, bi1, (short)0, im1, false, false);
  }
}

// out[c,t] = sum_r e^{+2pi i r c/16} * sum_k coeff[16k+r] * sig[16*(t+512-k) - r]
__global__ __launch_bounds__(256, 1)
void wdm_polyphase_wmma(const float* __restrict__ sig_re,
                        const float* __restrict__ sig_im,
                        const float* __restrict__ coeff,
                        float* __restrict__ out,
                        int sig_len, int out_t) {
  __shared__ __attribute__((aligned(16))) float s_coeff[16384];
  __shared__ __attribute__((aligned(16))) float s_sig[2 * SIG_STAGE];

  const int tid  = (int)threadIdx.x;
  const int wave = tid >> 5;
  const int lane = tid & 31;
  const int b    = lane & 15;   // WMMA M index (A) / N index (B) for this lane
  const int hl   = lane >> 4;   // half-wave select for K striping

  const int t0 = (int)blockIdx.x * TILE_T;
  const int j0 = t0 + 512;

  // sig element range needed: [16*(j0-1024)-15, 16*(j0+255)] inclusive
  const long long s0 = 16LL * (long long)(j0 - HALO_M) - 15;
  const long long s1 = s0 + SIG_STAGE;
  const long long c0 = s0 < 0 ? 0 : s0;
  const long long c1 = s1 > (long long)sig_len ? (long long)sig_len : s1;

  if (c0 != s0 || c1 != s1) {   // uniform per block: zero-fill at edges only
    float4 z = make_float4(0.f, 0.f, 0.f, 0.f);
    float4* p = (float4*)s_sig;
    for (int i = tid; i < (2 * SIG_STAGE) / 4; i += 256) p[i] = z;
    __syncthreads();
  }

  if (tid == 0) {               // one wave issues the TDM DMAs (EXEC ignored)
    __builtin_prefetch(coeff, 0, 0);
    unsigned lds_sig = (unsigned)(unsigned long long)(uintptr_t)s_sig;
    unsigned lds_cf  = (unsigned)(unsigned long long)(uintptr_t)s_coeff;
    tdm_load_1d(lds_cf, (unsigned long long)(uintptr_t)coeff, 16384u);
    if (c1 > c0) {
      unsigned nelem = (unsigned)(c1 - c0);
      unsigned shift = (unsigned)(c0 - s0) * 4u;
      tdm_load_1d(lds_sig + shift,
                  (unsigned long long)(uintptr_t)(sig_re + c0), nelem);
      tdm_load_1d(lds_sig + (unsigned)SIG_STAGE * 4u + shift,
                  (unsigned long long)(uintptr_t)(sig_im + c0), nelem);
    }
    __builtin_amdgcn_s_wait_tensorcnt(0);
  }
  __syncthreads();

  // ---- FIR: per wave 2 rows; O[b,a] = sum_e A_e[b,d] * X_r[d, a+e] ----
  v8f acc_re0 = (v8f)0.f, acc_im0 = (v8f)0.f;
  v8f acc_re1 = (v8f)0.f, acc_im1 = (v8f)0.f;
  const int r0 = wave << 1;
  const int r1 = r0 | 1;
  const int bB0 = (b << 8) + 15 - r1;    // B base at eidx=0 (even float index)

  // peeled boundary blocks (only places the tap mask matters)
  fir_step<0>(s_coeff, s_sig, 1024, bB0, b, hl, r0,
              acc_re0, acc_im0, acc_re1, acc_im1);
  #pragma unroll 1
  for (int eidx = 1; eidx < 64; ++eidx) {
    fir_step<1>(s_coeff, s_sig, 1024 - (eidx << 4), bB0 + (eidx << 8), b, hl, r0,
                acc_re0, acc_im0, acc_re1, acc_im1);
  }
  fir_step<2>(s_coeff, s_sig, 0, bB0 + (64 << 8), b, hl, r0,
              acc_re0, acc_im0, acc_re1, acc_im1);

  // ---- exchange filtered tile F[r, tloc] through LDS (overlay sig buffer) ----
  __syncthreads();
  {
    const int fbase = ((lane & 15) << 4) + (hl << 3);  // 8 consecutive floats
    *(float4*)&s_sig[r0 * 256 + fbase] =
        make_float4(acc_re0[0], acc_re0[1], acc_re0[2], acc_re0[3]);
    *(float4*)&s_sig[r0 * 256 + fbase + 4] =
        make_float4(acc_re0[4], acc_re0[5], acc_re0[6], acc_re0[7]);
    *(float4*)&s_sig[4096 + r0 * 256 + fbase] =
        make_float4(acc_im0[0], acc_im0[1], acc_im0[2], acc_im0[3]);
    *(float4*)&s_sig[4096 + r0 * 256 + fbase + 4] =
        make_float4(acc_im0[4], acc_im0[5], acc_im0[6], acc_im0[7]);
    *(float4*)&s_sig[r1 * 256 + fbase] =
        make_float4(acc_re1[0], acc_re1[1], acc_re1[2], acc_re1[3]);
    *(float4*)&s_sig[r1 * 256 + fbase + 4] =
        make_float4(acc_re1[4], acc_re1[5], acc_re1[6], acc_re1[7]);
    *(float4*)&s_sig[4096 + r1 * 256 + fbase] =
        make_float4(acc_im1[0], acc_im1[1], acc_im1[2], acc_im1[3]);
    *(float4*)&s_sig[4096 + r1 * 256 + fbase + 4] =
        make_float4(acc_im1[4], acc_im1[5], acc_im1[6], acc_im1[7]);
  }
  __syncthreads();

  // ---- 16-point IDFT across rows as complex 16x16x16 WMMA matmul ----
  const int cn = lane & 15;
  #pragma unroll 1
  for (int g = 0; g < 2; ++g) {
    const int tg = (wave << 1) | g;          // time group 0..15
    const int tloc = (tg << 4) + cn;
    v8f ore = (v8f)0.f, oim = (v8f)0.f;
    #pragma unroll
    for (int chunk = 0; chunk < 4; ++chunk) {
      v2f ec, es, en, fre, fim;
      #pragma unroll
      for (int i = 0; i < 2; ++i) {
        const int r = (chunk << 2) + (hl << 1) + i;
        const float ang = (float)((cn * r) & 15) * 0.39269908169872414f; // 2pi/16
        ec[i] = __cosf(ang);
        es[i] = __sinf(ang);
        en[i] = -es[i];
        fre[i] = s_sig[r * 256 + tloc];
        fim[i] = s_sig[4096 + r * 256 + tloc];
      }
      ore = __builtin_amdgcn_wmma_f32_16x16x4_f32(false, ec, false, fre, (short)0, ore, false, false);
      ore = __builtin_amdgcn_wmma_f32_16x16x4_f32(false, en, false, fim, (short)0, ore, false, false);
      oim = __builtin_amdgcn_wmma_f32_16x16x4_f32(false, es, false, fre, (short)0, oim, false, false);
      oim = __builtin_amdgcn_wmma_f32_16x16x4_f32(false, ec, false, fim, (short)0, oim, false, false);
    }
    const int t = t0 + tloc;
    if (t < out_t) {
      #pragma unroll
      for (int v = 0; v < 8; ++v) {
        const int c = v + (hl << 3);
        const long long oidx = 2LL * ((long long)c * (long long)out_t + t);
        *(float2*)&out[oidx] = make_float2(ore[v], oim[v]);   // b64 store
      }
    }
  }
}

extern "C" void kernel_launch(void* const* d_in, const int* in_sizes, int n_in,
                              void* d_out, int out_size, void* d_ws, size_t ws_size,
                              hipStream_t stream) {
  (void)n_in; (void)out_size; (void)d_ws; (void)ws_size;
  const float* sr = (const float*)d_in[0];
  const float* si = (const float*)d_in[1];
  const float* cf = (const float*)d_in[2];
  float* out = (float*)d_out;
  const int sig_len = in_sizes[0];          // 8388608
  const int nper    = sig_len / 16;         // 524288
  const int out_t   = nper - 1;             // 524287 kept columns
  const int nblk    = (out_t + TILE_T - 1) / TILE_T;
  hipLaunchKernelGGL(wdm_polyphase_wmma, dim3(nblk), dim3(256), 0, stream,
                     sr, si, cf, out, sig_len, out_t);
}